// GINModel_66022237274357
// MI455X (gfx1250) — compile-verified
//
#include <hip/hip_runtime.h>

// GIN model for MI455X (gfx1250): fp32 WMMA (v_wmma_f32_16x16x4_f32) GEMMs with
// compile-time K/NV, clamped (branch-free) tile loads and pre-transposed
// weights; atomic scatter-add aggregation; fused bias+BN+LeakyReLU epilogues.
//
// Roofline: workload is memory/atomic bound (edge aggregation ~2.4 GB of
// L2-resident traffic per layer vs. only ~39 GFLOP of GEMM total), so we keep
// full fp32 precision via V_WMMA_F32_16X16X4_F32 rather than downconverting.

#define NNODES 100000
#define NEDGES 1600000
#define CH     128      // IN == H == 128
#define COUT   10
#define NGRAPH 128
#define BN_EPS 1e-5f

typedef float v8f __attribute__((ext_vector_type(8)));
typedef float v2f __attribute__((ext_vector_type(2)));

__global__ void zero_kernel(float* __restrict__ p, long n) {
  long i = (long)blockIdx.x * blockDim.x + threadIdx.x;
  if (i < n) p[i] = 0.0f;
}

// One wave per edge: lane l handles channels [4l, 4l+4).
__global__ void scatter_add_kernel(const float* __restrict__ x,
                                   const int* __restrict__ ei,   // [2, E] flattened
                                   float* __restrict__ aggr, int E) {
  long tid  = (long)blockIdx.x * blockDim.x + threadIdx.x;
  int  edge = (int)(tid >> 5);
  int  lane = (int)(tid & 31);
  if (edge >= E) return;
  int src = ei[edge];
  int dst = ei[E + edge];
  const float4 v = *(const float4*)(x + (size_t)src * CH + lane * 4);
  float* base = aggr + (size_t)dst * CH + lane * 4;
  atomicAdd(base + 0, v.x);
  atomicAdd(base + 1, v.y);
  atomicAdd(base + 2, v.z);
  atomicAdd(base + 3, v.w);
}

// In-place:  aggr = (1 + eps) * x + aggr    (float4 per thread)
__global__ void compose_kernel(const float* __restrict__ x,
                               const float* __restrict__ epsp,
                               float* __restrict__ aggr, long n4) {
  long i = (long)blockIdx.x * blockDim.x + threadIdx.x;
  if (i >= n4) return;
  const float alpha = 1.0f + epsp[0];
  float4 xv = ((const float4*)x)[i];
  float4 av = ((float4*)aggr)[i];
  av.x += alpha * xv.x;
  av.y += alpha * xv.y;
  av.z += alpha * xv.z;
  av.w += alpha * xv.w;
  ((float4*)aggr)[i] = av;
}

// Wt[n*K + k] = W[k*Nv + n]
__global__ void transpose_kernel(const float* __restrict__ W,
                                 float* __restrict__ Wt, int K, int Nv) {
  long tid = (long)blockIdx.x * blockDim.x + threadIdx.x;
  if (tid >= (long)K * Nv) return;
  int k = (int)(tid / Nv);
  int n = (int)(tid % Nv);
  Wt[(size_t)n * K + k] = W[tid];
}

// Global add pool: one thread per (node, channel).
__global__ void pool_kernel(const float* __restrict__ x,
                            const int* __restrict__ batch,
                            float* __restrict__ pooled, int N) {
  long tid = (long)blockIdx.x * blockDim.x + threadIdx.x;
  int i = (int)(tid >> 7);      // CH == 128
  int c = (int)(tid & 127);
  if (i >= N) return;
  atomicAdd(&pooled[(size_t)batch[i] * CH + c], x[(size_t)i * CH + c]);
}

// Out[M,NV] = act( BN( A @ Wt^T + bias ) )
//   A:[M,K] row-major, Wt:[NV,K] row-major (pre-transposed weight).
// One wave = one 16x16 output tile via V_WMMA_F32_16X16X4_F32; 8 waves/block
// cover 8 consecutive row tiles at one column tile. All loads use clamped
// addresses (out-of-range rows/cols compute garbage that is never stored),
// so the K-loop is branch-free and EXEC is all-ones at every WMMA.
// K/NV are compile-time: constant trip count (no remainder loop), immediate
// strides in addressing.
template <int K, int NV>
__global__ __launch_bounds__(256)
void gemm_wmma_kernel(const float* __restrict__ A,
                      const float* __restrict__ Wt,
                      const float* __restrict__ bias,
                      const float* __restrict__ gamma,
                      const float* __restrict__ beta,
                      const float* __restrict__ mean,
                      const float* __restrict__ var,
                      float slope, int do_act,
                      float* __restrict__ Out, int M) {
  const int wave = threadIdx.x >> 5;
  const int lane = threadIdx.x & 31;
  const int half = lane >> 4;          // selects K pair within the 16x4 A tile
  const int l16  = lane & 15;

  const int rowBase = (blockIdx.x * 8 + wave) * 16;
  const int cc  = blockIdx.y * 16 + l16;         // output column of this lane
  const int ccl = (NV % 16 == 0) ? cc : min(cc, NV - 1);  // clamp for load
  const int ar  = rowBase + l16;                 // A row this lane supplies
  const int arc = min(ar, M - 1);

  // Running pointers: per WMMA step, lane consumes A[m][ka..ka+1] and
  // Wt[n][ka..ka+1] with ka = k0 + half*2  (ISA f32 16x4 operand layout).
  const float2* Ap = (const float2*)(A  + (size_t)arc * K + half * 2);
  const float2* Bp = (const float2*)(Wt + (size_t)ccl * K + half * 2);

  v8f acc = {0.f, 0.f, 0.f, 0.f, 0.f, 0.f, 0.f, 0.f};

#pragma unroll 8
  for (int k0 = 0; k0 < K; k0 += 4) {
    const float2 av = *Ap;
    const float2 bv = *Bp;
    v2f a; a.x = av.x; a.y = av.y;
    v2f b; b.x = bv.x; b.y = bv.y;
    acc = __builtin_amdgcn_wmma_f32_16x16x4_f32(
        /*neg_a=*/false, a, /*neg_b=*/false, b,
        /*c_mod=*/(short)0, acc, /*reuse_a=*/false, /*reuse_b=*/false);
    Ap += 2;   // advance 4 floats
    Bp += 2;
  }

  // Epilogue: bias -> BN -> LeakyReLU (per output column)
  float bv = bias[ccl];
  float scale = 1.0f, shift = 0.0f;
  if (gamma) {
    float s = gamma[ccl] * rsqrtf(var[ccl] + BN_EPS);
    scale = s;
    shift = beta[ccl] - mean[ccl] * s;
  }

  const bool ccOK = (NV % 16 == 0) || (cc < NV);
#pragma unroll
  for (int r = 0; r < 8; ++r) {
    const int m = rowBase + r + half * 8;        // C/D layout: lanes16-31 -> M+8
    float v = acc[r] + bv;
    v = v * scale + shift;
    if (do_act) v = (v > 0.0f) ? v : v * slope;
    if (m < M && ccOK) Out[(size_t)m * NV + cc] = v;
  }
}

extern "C" void kernel_launch(void* const* d_in, const int* in_sizes, int n_in,
                              void* d_out, int out_size, void* d_ws, size_t ws_size,
                              hipStream_t stream) {
  const int N = NNODES, E = NEDGES, G = NGRAPH;

  const float* x0    = (const float*)d_in[0];
  const int*   ei    = (const int*)d_in[1];
  const int*   batch = (const int*)d_in[2];
  auto F = [&](int i) { return (const float*)d_in[i]; };

  // Workspace (floats):
  //   bufX[N*128] | aggr[N*128] | h[N*256] | pooled[G*128] | h1[G*128] | wT[256*128]
  float* ws     = (float*)d_ws;
  float* bufX   = ws;
  float* aggr   = bufX   + (size_t)N * CH;
  float* hbuf   = aggr   + (size_t)N * CH;
  float* pooled = hbuf   + (size_t)N * (2 * CH);
  float* h1     = pooled + (size_t)G * CH;
  float* wT     = h1     + (size_t)G * CH;       // up to 2*CH*CH floats

  dim3 blk(256);
  const int rowTiles = (N + 15) / 16;            // 6250
  const dim3 gridG1((rowTiles + 7) / 8, (2 * CH) / 16);
  const dim3 gridG2((rowTiles + 7) / 8, CH / 16);

  auto transpose = [&](const float* W, int K, int Nv) {
    long n = (long)K * Nv;
    transpose_kernel<<<(int)((n + 255) / 256), blk, 0, stream>>>(W, wT, K, Nv);
  };

  const float* xin = x0;
  for (int l = 0; l < 3; ++l) {
    const int p = 3 + 13 * l;
    const float* eps = F(p + 0);
    const float* W1  = F(p + 1), *b1 = F(p + 2);
    const float* g1  = F(p + 3), *be1 = F(p + 4), *mu1 = F(p + 5), *va1 = F(p + 6);
    const float* W2  = F(p + 7), *b2 = F(p + 8);
    const float* g2  = F(p + 9), *be2 = F(p + 10), *mu2 = F(p + 11), *va2 = F(p + 12);

    // aggr = segment_sum(x[src], dst);  then aggr = (1+eps)*x + aggr
    const long na = (long)N * CH;
    zero_kernel<<<(int)((na + 255) / 256), blk, 0, stream>>>(aggr, na);
    const long nt = (long)E * 32;
    scatter_add_kernel<<<(int)((nt + 255) / 256), blk, 0, stream>>>(xin, ei, aggr, E);
    compose_kernel<<<(int)((na / 4 + 255) / 256), blk, 0, stream>>>(xin, eps, aggr, na / 4);

    // h = LeakyReLU(BN( aggr @ W1 + b1 ))
    transpose(W1, CH, 2 * CH);
    gemm_wmma_kernel<CH, 2 * CH><<<gridG1, blk, 0, stream>>>(
        aggr, wT, b1, g1, be1, mu1, va1, 0.01f, 1, hbuf, N);

    // x = LeakyReLU(BN( h @ W2 + b2 ))
    transpose(W2, 2 * CH, CH);
    gemm_wmma_kernel<2 * CH, CH><<<gridG2, blk, 0, stream>>>(
        hbuf, wT, b2, g2, be2, mu2, va2, 0.01f, 1, bufX, N);

    xin = bufX;
  }

  // pooled = segment_sum(x, batch)
  const long np = (long)G * CH;
  zero_kernel<<<(int)((np + 255) / 256), blk, 0, stream>>>(pooled, np);
  const long ntp = (long)N * CH;
  pool_kernel<<<(int)((ntp + 255) / 256), blk, 0, stream>>>(xin, batch, pooled, N);

  // h1 = ReLU(BN( pooled @ lin1_W + lin1_b ))
  {
    transpose(F(42), CH, CH);
    dim3 grid(1, CH / 16);
    gemm_wmma_kernel<CH, CH><<<grid, blk, 0, stream>>>(
        pooled, wT, F(43), F(44), F(45), F(46), F(47), 0.0f, 1, h1, G);
  }
  // out = h1 @ lin2_W + lin2_b   (dropout is identity in eval)
  {
    transpose(F(48), CH, COUT);
    dim3 grid(1, 1);
    gemm_wmma_kernel<CH, COUT><<<grid, blk, 0, stream>>>(
        h1, wT, F(49), nullptr, nullptr, nullptr, nullptr, 0.0f, 0,
        (float*)d_out, G);
  }
}